// MutualInformationLoss_32839319945330
// MI455X (gfx1250) — compile-verified
//
#include <hip/hip_runtime.h>
#include <math.h>

// Mutual-information loss: two 30-bin histograms over [-4,4], analytic
// collapse of the 30x30 double sum. Memory-bound: 256 MiB @ 23.3 TB/s ~ 11us.

#define BINS   30
#define NCOPY  32   // one LDS histogram copy per wave32 lane -> 0 bank conflicts
#define TRASH  31   // bin for out-of-range values (ignored by histc)

typedef __attribute__((ext_vector_type(4))) float v4f;
typedef __attribute__((ext_vector_type(2))) float v2f;
typedef __attribute__((ext_vector_type(8))) float v8f;

__device__ __forceinline__ int bin_of(float v) {
    // width = 8/30 -> 1/width = 3.75 ; v==+4 lands in bin 29 (torch.histc rule)
    int idx = (int)floorf((v + 4.0f) * 3.75f);
    idx = idx < 0 ? 0 : (idx > BINS - 1 ? BINS - 1 : idx);
    bool valid = (v >= -4.0f) && (v <= 4.0f);
    return valid ? idx : TRASH;
}

__global__ void __launch_bounds__(256)
mi_hist_kernel(const float* __restrict__ x, const float* __restrict__ y,
               int nx, int ny, float* __restrict__ partials) {
    // hist[type][bin][copy]: element address mod 64 banks = (bin*32 + copy) % 64;
    // two distinct lanes of a wave32 always differ mod 32 -> distinct banks ->
    // conflict-free ds_add_u32 for ANY bin pattern (normal data hammers center bins).
    __shared__ unsigned hist[2][32][NCOPY];
    const int tid = threadIdx.x;
    const int cpy = tid & 31;

    for (int i = tid; i < 2 * 32 * NCOPY; i += blockDim.x)
        ((unsigned*)hist)[i] = 0u;
    __syncthreads();

    const long long gtid   = (long long)blockIdx.x * blockDim.x + tid;
    const long long stride = (long long)gridDim.x * blockDim.x;

    // ---- histogram x ----
    {
        const v4f* x4 = (const v4f*)x;
        const long long n4 = (long long)nx >> 2;
        for (long long i = gtid; i < n4; i += stride) {
            __builtin_prefetch((const char*)(x4 + i + 2 * stride), 0, 1); // global_prefetch_b8
            v4f v = __builtin_nontemporal_load(x4 + i);                   // b128, th:NT
            atomicAdd(&hist[0][bin_of(v.x)][cpy], 1u);
            atomicAdd(&hist[0][bin_of(v.y)][cpy], 1u);
            atomicAdd(&hist[0][bin_of(v.z)][cpy], 1u);
            atomicAdd(&hist[0][bin_of(v.w)][cpy], 1u);
        }
        const long long base = n4 << 2;
        if (base + gtid < (long long)nx)
            atomicAdd(&hist[0][bin_of(x[base + gtid])][cpy], 1u);
    }
    // ---- histogram y ----
    {
        const v4f* y4 = (const v4f*)y;
        const long long n4 = (long long)ny >> 2;
        for (long long i = gtid; i < n4; i += stride) {
            __builtin_prefetch((const char*)(y4 + i + 2 * stride), 0, 1);
            v4f v = __builtin_nontemporal_load(y4 + i);
            atomicAdd(&hist[1][bin_of(v.x)][cpy], 1u);
            atomicAdd(&hist[1][bin_of(v.y)][cpy], 1u);
            atomicAdd(&hist[1][bin_of(v.z)][cpy], 1u);
            atomicAdd(&hist[1][bin_of(v.w)][cpy], 1u);
        }
        const long long base = n4 << 2;
        if (base + gtid < (long long)ny)
            atomicAdd(&hist[1][bin_of(y[base + gtid])][cpy], 1u);
    }
    __syncthreads();

    // Collapse the 32 copies; every WG fully rewrites its 64-float slot
    // (slots 30,31,62,63 written as zero) -> replay-deterministic, no memset.
    if (tid < 64) {
        const int t = tid >> 5, b = tid & 31;
        unsigned s = 0;
        #pragma unroll
        for (int k = 0; k < NCOPY; ++k) s += hist[t][b][k];
        partials[(long long)blockIdx.x * 64 + tid] = (b < BINS) ? (float)s : 0.0f;
    }
}

// One wave32. Reduce P[nwg][64] column-wise with V_WMMA_F32_16X16X4_F32:
// A = ones(16x4) => D[m][n] = sum_k B[k][n] + C[m][n]; iterate K over row
// quads (guard-free steady state; nwg is a multiple of 4 from the launcher),
// 4 column groups of 16. Then the O(30) analytic MI epilogue.
__global__ void __launch_bounds__(32)
mi_finalize_kernel(const float* __restrict__ P, int nwg, float* __restrict__ out) {
    __shared__ float colsum[64];
    const int lane = threadIdx.x;          // 0..31; EXEC all ones at every WMMA

    v2f a; a.x = 1.0f; a.y = 1.0f;         // A-matrix: all ones (16x4 f32)
    const int colbase = lane & 15;
    const int rsel    = (lane >> 4) << 1;  // lanes 0-15 feed K=0,1 ; lanes 16-31 feed K=2,3

    for (int grp = 0; grp < 4; ++grp) {
        const int col = grp * 16 + colbase;
        v8f c = {};
        int r0 = 0;
        // Guard-free steady state: unconditional b32 loads, pipelineable.
        #pragma unroll 2
        for (; r0 + 4 <= nwg; r0 += 4) {
            const float* row = P + (long long)(r0 + rsel) * 64 + col;
            v2f b; b.x = row[0]; b.y = row[64];
            c = __builtin_amdgcn_wmma_f32_16x16x4_f32(
                    false, a, false, b, (short)0, c, false, false);
        }
        // Wave-uniform tail (never taken when nwg % 4 == 0).
        if (r0 < nwg) {
            const int r = r0 + rsel;
            v2f b;
            b.x = (r     < nwg) ? P[(long long)r       * 64 + col] : 0.0f;
            b.y = (r + 1 < nwg) ? P[(long long)(r + 1) * 64 + col] : 0.0f;
            c = __builtin_amdgcn_wmma_f32_16x16x4_f32(
                    false, a, false, b, (short)0, c, false, false);
        }
        // D row 0 (C/D VGPR0, lanes 0..15) holds the column sums for this group.
        if (lane < 16) colsum[col] = c[0];
        __syncthreads();
    }

    if (lane == 0) {
        float hx[BINS], hy[BINS], Sx = 0.0f, Sy = 0.0f;
        for (int b = 0; b < BINS; ++b) {
            hx[b] = colsum[b];
            hy[b] = colsum[32 + b];
            Sx += hx[b];
            Sy += hy[b];
        }
        const float Sj = Sx + Sy;
        float s_jlnj = 0.0f, s_lnpx = 0.0f, s_jlnpy = 0.0f, s_jp = 0.0f;
        for (int b = 0; b < BINS; ++b) {
            const float jp = (hx[b] + hy[b]) / Sj;
            const float px = hx[b] / Sx;
            const float py = hy[b] / Sy;
            s_jp    += jp;
            s_jlnj  += jp * logf(jp);
            s_lnpx  += logf(px);
            s_jlnpy += jp * logf(py);
        }
        const float mi = (float)BINS * s_jlnj - s_jp * s_lnpx - (float)BINS * s_jlnpy;
        out[0] = -mi;
    }
}

extern "C" void kernel_launch(void* const* d_in, const int* in_sizes, int n_in,
                              void* d_out, int out_size, void* d_ws, size_t ws_size,
                              hipStream_t stream) {
    const float* x = (const float*)d_in[0];
    const float* y = (const float*)d_in[1];
    const int nx = in_sizes[0];
    const int ny = (n_in > 1) ? in_sizes[1] : 0;

    float* partials = (float*)d_ws;

    int nwg = 1024;                                   // 1024 WGs x 8 waves, 256 KiB scratch
    const size_t slot = 64 * sizeof(float);
    if (ws_size < (size_t)nwg * slot) {
        nwg = (int)(ws_size / slot);
        nwg &= ~3;                                    // keep the WMMA K-loop guard-free
        if (nwg < 4) nwg = 4;                         // ws_size >= 1 KiB assumed
    }

    mi_hist_kernel<<<nwg, 256, 0, stream>>>(x, y, nx, ny, partials);
    mi_finalize_kernel<<<1, 32, 0, stream>>>(partials, nwg, (float*)d_out);
}